// SemanticModule_8194797601369
// MI455X (gfx1250) — compile-verified
//
#include <hip/hip_runtime.h>

// ---------------------------------------------------------------------------
// SemanticModule GNN for MI455X (gfx1250, wave32).
// Strategy: aggregate-then-transform (segment_sum commutes with the linear
// layer), so edge scatter traffic is over C_in channels, and all dense math
// runs through V_WMMA_F32_16X16X4_F32 with fp32 accumulation.
// All feature/agg buffers (< 80 MB) live in d_ws and fit in the 192 MB L2,
// so gathers and f32 atomic scatters are L2-resident.
// ---------------------------------------------------------------------------

typedef float v2f __attribute__((ext_vector_type(2)));
typedef float v8f __attribute__((ext_vector_type(8)));

__global__ void zero_f32_kernel(float* __restrict__ p, int n) {
  int i = blockIdx.x * blockDim.x + threadIdx.x;
  if (i < n) p[i] = 0.0f;
}

// deg[di[e]] += 1  (float degree accumulation for the mean relation)
__global__ void deg_kernel(const int* __restrict__ di, float* __restrict__ deg, int E) {
  int e = blockIdx.x * blockDim.x + threadIdx.x;
  if (e < E) unsafeAtomicAdd(&deg[di[e]], 1.0f);
}

__global__ void invdeg_kernel(float* __restrict__ deg, int n) {
  int i = blockIdx.x * blockDim.x + threadIdx.x;
  if (i < n) deg[i] = 1.0f / fmaxf(deg[i], 1.0f);
}

// agg[dst[e]*C + c..c+V) += feat[src[e]*C + c..c+V)
// C compile-time: /CPT lowers to a shift for C=32; V=4 uses global_load_b128
// gathers so each lane moves 16B and src/dst index loads drop 4x.
// Atomic f32 adds (global_atomic_add_f32) remain the irreducible cost.
template <int C>
__global__ void scatter_add_kernel(const float* __restrict__ feat,
                                   const int* __restrict__ src,
                                   const int* __restrict__ dst,
                                   float* __restrict__ agg,
                                   int E) {
  constexpr int V = (C % 4 == 0) ? 4 : 1;   // channels per lane
  constexpr int CPT = C / V;                // lanes per edge
  int gid = blockIdx.x * blockDim.x + threadIdx.x;
  if (gid >= E * CPT) return;
  int e = gid / CPT;                        // shift when CPT is a power of two
  int c = (gid - e * CPT) * V;
  int s = src[e];
  int d = dst[e];
  if (V == 4) {
    const float4 v = *(const float4*)(feat + s * C + c);
    unsafeAtomicAdd(&agg[d * C + c + 0], v.x);
    unsafeAtomicAdd(&agg[d * C + c + 1], v.y);
    unsafeAtomicAdd(&agg[d * C + c + 2], v.z);
    unsafeAtomicAdd(&agg[d * C + c + 3], v.w);
  } else {
    unsafeAtomicAdd(&agg[d * C + c], feat[s * C + c]);
  }
}

// Fused conv: out = maybe_relu( A0@W0 + (inv_deg ⊙ A1)@W1 + b0 + b1 )
//                   [+ resid elementwise] [+ projA@Wp]
// One wave per 16-row block; all OUT/16 column tiles per wave.
// V_WMMA_F32_16X16X4_F32 lane layout:
//   A 16x4: lane r=lane&15 -> row M=r; half=lane>>4 selects K pair {0,1}/{2,3}
//   B 4x16: same indexing transposed (row K, col N=r)
//   C/D v8f: element (M = j + 8*half, N = r) in acc[j]
template <int K, int OUT>
__global__ void conv_wmma_kernel(const float* __restrict__ A0,
                                 const float* __restrict__ W0,
                                 const float* __restrict__ b0,
                                 const float* __restrict__ A1,
                                 const float* __restrict__ W1,
                                 const float* __restrict__ b1,
                                 const float* __restrict__ inv_deg,
                                 const float* __restrict__ resid,
                                 const float* __restrict__ projA,
                                 const float* __restrict__ Wp,
                                 float* __restrict__ out,
                                 int N, int do_relu) {
  int wave = (blockIdx.x * blockDim.x + threadIdx.x) >> 5;
  int lane = threadIdx.x & 31;
  int m0 = wave << 4;
  if (m0 >= N) return;  // wave-uniform exit; surviving waves keep EXEC all-1s

  const int r = lane & 15;
  const int half = lane >> 4;
  const int row = m0 + r;
  const float s = (A1 != nullptr) ? inv_deg[row] : 1.0f;

  constexpr int NT = OUT / 16;
#pragma unroll
  for (int t = 0; t < NT; ++t) {
    const int n0 = t << 4;
    v8f acc = {};

    // A0 @ W0
#pragma unroll
    for (int k0 = 0; k0 < K; k0 += 4) {
      const int kk = k0 + 2 * half;
      v2f a, b;
      a.x = (kk     < K) ? A0[row * K + kk]           : 0.0f;
      a.y = (kk + 1 < K) ? A0[row * K + kk + 1]       : 0.0f;
      b.x = (kk     < K) ? W0[kk * OUT + n0 + r]      : 0.0f;
      b.y = (kk + 1 < K) ? W0[(kk + 1) * OUT + n0 + r]: 0.0f;
      acc = __builtin_amdgcn_wmma_f32_16x16x4_f32(false, a, false, b,
                                                  (short)0, acc, false, false);
    }
    // (inv_deg * A1) @ W1  (mean relation, row scaling commutes with matmul)
    if (A1 != nullptr) {
#pragma unroll
      for (int k0 = 0; k0 < K; k0 += 4) {
        const int kk = k0 + 2 * half;
        v2f a, b;
        a.x = (kk     < K) ? A1[row * K + kk] * s           : 0.0f;
        a.y = (kk + 1 < K) ? A1[row * K + kk + 1] * s       : 0.0f;
        b.x = (kk     < K) ? W1[kk * OUT + n0 + r]          : 0.0f;
        b.y = (kk + 1 < K) ? W1[(kk + 1) * OUT + n0 + r]    : 0.0f;
        acc = __builtin_amdgcn_wmma_f32_16x16x4_f32(false, a, false, b,
                                                    (short)0, acc, false, false);
      }
    }

    float bias = 0.0f;
    if (b0) bias += b0[n0 + r];
    if (b1) bias += b1[n0 + r];

#pragma unroll
    for (int j = 0; j < 8; ++j) {
      float v = acc[j] + bias;
      if (do_relu) v = fmaxf(v, 0.0f);
      acc[j] = v;
    }

    if (resid != nullptr) {
#pragma unroll
      for (int j = 0; j < 8; ++j) {
        const int m = m0 + j + 8 * half;
        acc[j] += resid[m * OUT + n0 + r];
      }
    }

    // Projected shortcut: + projA @ Wp (post-ReLU), chained on the accumulator
    if (projA != nullptr) {
#pragma unroll
      for (int k0 = 0; k0 < K; k0 += 4) {
        const int kk = k0 + 2 * half;
        v2f a, b;
        a.x = (kk     < K) ? projA[row * K + kk]            : 0.0f;
        a.y = (kk + 1 < K) ? projA[row * K + kk + 1]        : 0.0f;
        b.x = (kk     < K) ? Wp[kk * OUT + n0 + r]          : 0.0f;
        b.y = (kk + 1 < K) ? Wp[(kk + 1) * OUT + n0 + r]    : 0.0f;
        acc = __builtin_amdgcn_wmma_f32_16x16x4_f32(false, a, false, b,
                                                    (short)0, acc, false, false);
      }
    }

#pragma unroll
    for (int j = 0; j < 8; ++j) {
      const int m = m0 + j + 8 * half;
      out[m * OUT + n0 + r] = acc[j];
    }
  }
}

static inline unsigned nblk(long long n, int bs) {
  return (unsigned)((n + bs - 1) / bs);
}

extern "C" void kernel_launch(void* const* d_in, const int* in_sizes, int n_in,
                              void* d_out, int out_size, void* d_ws, size_t ws_size,
                              hipStream_t stream) {
  (void)n_in; (void)out_size; (void)ws_size;
  const float* x     = (const float*)d_in[0];
  const int*   e_t   = (const int*)d_in[1];
  const int*   e_i   = (const int*)d_in[2];
  const float* W_t0  = (const float*)d_in[3];
  const float* b_t0  = (const float*)d_in[4];
  const float* W_i0  = (const float*)d_in[5];
  const float* b_i0  = (const float*)d_in[6];
  const float* W_t   = (const float*)d_in[7];   // [3,32,32]
  const float* b_t   = (const float*)d_in[8];   // [3,32]
  const float* W_i   = (const float*)d_in[9];
  const float* b_i   = (const float*)d_in[10];
  const float* W_t4  = (const float*)d_in[11];  // [32,64]
  const float* b_t4  = (const float*)d_in[12];
  const float* W_i4  = (const float*)d_in[13];
  const float* b_i4  = (const float*)d_in[14];
  const float* W_res = (const float*)d_in[15];  // [32,64]
  const float* W_mlp = (const float*)d_in[16];  // [64,32]
  const float* b_mlp = (const float*)d_in[17];

  const int N  = in_sizes[0] / 6;
  const int ET = in_sizes[1] / 2;
  const int EI = in_sizes[2] / 2;
  const int* st = e_t;       const int* dt = e_t + ET;
  const int* si = e_i;       const int* di = e_i + EI;

  // Workspace layout (floats): inv_deg | aggT(N*32) | aggI(N*32) | hA(N*64) | hB(N*64)
  float* ws      = (float*)d_ws;
  float* inv_deg = ws;
  float* aggT    = inv_deg + N;
  float* aggI    = aggT + (size_t)N * 32;
  float* hA      = aggI + (size_t)N * 32;
  float* hB      = hA + (size_t)N * 64;

  const int BS = 256;
  const unsigned cb = nblk((long long)((N + 15) / 16) * 32, BS);  // conv waves

  // ---- degrees for the mean relation ----
  zero_f32_kernel<<<nblk(N, BS), BS, 0, stream>>>(inv_deg, N);
  deg_kernel<<<nblk(EI, BS), BS, 0, stream>>>(di, inv_deg, EI);
  invdeg_kernel<<<nblk(N, BS), BS, 0, stream>>>(inv_deg, N);

  // ---- layer 0: x(6) -> h(32), ReLU, no residual ----
  // one zero launch covers both (contiguous) agg buffers
  zero_f32_kernel<<<nblk((long long)N * 64, BS), BS, 0, stream>>>(aggT, N * 64);
  scatter_add_kernel<6><<<nblk((long long)ET * 6, BS), BS, 0, stream>>>(x, st, dt, aggT, ET);
  scatter_add_kernel<6><<<nblk((long long)EI * 6, BS), BS, 0, stream>>>(x, si, di, aggI, EI);
  conv_wmma_kernel<6, 32><<<cb, BS, 0, stream>>>(
      aggT, W_t0, b_t0, aggI, W_i0, b_i0, inv_deg,
      nullptr, nullptr, nullptr, hA, N, 1);

  // ---- layers 1-3: h(32) -> h(32), ReLU + identity residual ----
  const float* hin = hA;
  float* hout = hB;
  for (int l = 0; l < 3; ++l) {
    zero_f32_kernel<<<nblk((long long)N * 64, BS), BS, 0, stream>>>(aggT, N * 64);
    scatter_add_kernel<32><<<nblk((long long)ET * 8, BS), BS, 0, stream>>>(hin, st, dt, aggT, ET);
    scatter_add_kernel<32><<<nblk((long long)EI * 8, BS), BS, 0, stream>>>(hin, si, di, aggI, EI);
    conv_wmma_kernel<32, 32><<<cb, BS, 0, stream>>>(
        aggT, W_t + l * 32 * 32, b_t + l * 32,
        aggI, W_i + l * 32 * 32, b_i + l * 32, inv_deg,
        hin /*resid*/, nullptr, nullptr, hout, N, 1);
    const float* tmp = hin; hin = hout; hout = (float*)tmp;
  }

  // ---- layer 4: h(32) -> h(64), ReLU + projected shortcut h@W_res ----
  zero_f32_kernel<<<nblk((long long)N * 64, BS), BS, 0, stream>>>(aggT, N * 64);
  scatter_add_kernel<32><<<nblk((long long)ET * 8, BS), BS, 0, stream>>>(hin, st, dt, aggT, ET);
  scatter_add_kernel<32><<<nblk((long long)EI * 8, BS), BS, 0, stream>>>(hin, si, di, aggI, EI);
  conv_wmma_kernel<32, 64><<<cb, BS, 0, stream>>>(
      aggT, W_t4, b_t4, aggI, W_i4, b_i4, inv_deg,
      nullptr, hin /*projA*/, W_res, hout, N, 1);

  // ---- MLP head: h(64) @ W_mlp + b_mlp -> out(32), no ReLU ----
  conv_wmma_kernel<64, 32><<<cb, BS, 0, stream>>>(
      hout, W_mlp, b_mlp, nullptr, nullptr, nullptr, nullptr,
      nullptr, nullptr, nullptr, (float*)d_out, N, 0);
}